// Ultimus_5927054869140
// MI455X (gfx1250) — compile-verified
//
#include <hip/hip_runtime.h>

typedef __attribute__((ext_vector_type(2))) float v2f;
typedef __attribute__((ext_vector_type(8))) float v8f;

#define HH 4096
#define CIN 48
#define CBOT 8
#define NWAVE 8
#define ATT_SCALE 0.35355339059327373f  // 1/sqrt(8)

__global__ __launch_bounds__(256)
void ultimus_fused(const float* __restrict__ x,
                   const float* __restrict__ w_down,
                   const float* __restrict__ w_up,
                   float* __restrict__ out) {
    // 128 KB kqv tile held on-chip for the whole batch: uses CDNA5's 320KB WGP LDS.
    __shared__ float s_kqv[CBOT][HH];
    __shared__ float s_gp[NWAVE][40];
    __shared__ float s_G[8][8];
    __shared__ float s_am[8][8];
    __shared__ float s_W2[CIN][8];

    const int t    = threadIdx.x;
    const int wave = t >> 5;
    const int lane = t & 31;
    const int col  = lane & 15;          // N index within a 16-wide tile / M row for A
    const int krow = (lane >> 4) << 1;   // K base held by this half-wave (0 or 2)
    const int b    = blockIdx.x;

    const float* xb   = x   + (size_t)b * CIN * HH;
    float*       outb = out + (size_t)b * CIN * HH;

    // ================= Phase 1: kqv = w_down @ x   (8 x H) =================
    // A = w_down zero-padded to 16 rows; preload all 12 K-chunks (24 VGPRs).
    v2f areg[12];
#pragma unroll
    for (int kc = 0; kc < 12; ++kc) {
        float ax = 0.f, ay = 0.f;
        if (col < CBOT) {
            ax = w_down[col * CIN + kc * 4 + krow];
            ay = w_down[col * CIN + kc * 4 + krow + 1];
        }
        areg[kc].x = ax;
        areg[kc].y = ay;
    }

    for (int ht = wave; ht < HH / 16; ht += NWAVE) {
        const int h0 = ht * 16;
        v8f acc = {};
#pragma unroll
        for (int kc = 0; kc < 12; ++kc) {
            const int c0 = kc * 4 + krow;
            v2f bv;
            bv.x = xb[(size_t)c0 * HH + h0 + col];
            bv.y = xb[(size_t)(c0 + 1) * HH + h0 + col];
            acc = __builtin_amdgcn_wmma_f32_16x16x4_f32(
                false, areg[kc], false, bv, (short)0, acc, false, false);
        }
        // D rows 0..7 (lanes 0-15) are the valid kqv rows; rows 8..15 are zero pad.
        if (lane < 16) {
#pragma unroll
            for (int r = 0; r < 8; ++r) s_kqv[r][h0 + lane] = acc[r];
        }
    }
    __syncthreads();

    // ============ Phase 2: G = kqv kqv^T (8x8), softmax, W2 fold ============
    float g[36];
#pragma unroll
    for (int i = 0; i < 36; ++i) g[i] = 0.f;
    for (int h = t; h < HH; h += 256) {
        float v[8];
#pragma unroll
        for (int i = 0; i < 8; ++i) v[i] = s_kqv[i][h];
        int idx = 0;
#pragma unroll
        for (int i = 0; i < 8; ++i) {
#pragma unroll
            for (int j = i; j < 8; ++j) {
                g[idx] = fmaf(v[i], v[j], g[idx]);
                ++idx;
            }
        }
    }
    // wave32 butterfly reduce, then cross-wave via LDS
#pragma unroll
    for (int i = 0; i < 36; ++i) {
#pragma unroll
        for (int off = 16; off > 0; off >>= 1)
            g[i] += __shfl_xor(g[i], off, 32);
    }
    if (lane == 0) {
#pragma unroll
        for (int i = 0; i < 36; ++i) s_gp[wave][i] = g[i];
    }
    __syncthreads();
    if (t < 36) {
        float s = 0.f;
#pragma unroll
        for (int w = 0; w < NWAVE; ++w) s += s_gp[w][t];
        int i = 0, base = 0;
        while (t >= base + (8 - i)) { base += 8 - i; ++i; }
        const int j = i + (t - base);
        s_G[i][j] = s;
        s_G[j][i] = s;
    }
    __syncthreads();
    if (t < 8) {
        float m = -1e30f;
#pragma unroll
        for (int j = 0; j < 8; ++j) m = fmaxf(m, s_G[t][j] * ATT_SCALE);
        float e[8], sum = 0.f;
#pragma unroll
        for (int j = 0; j < 8; ++j) {
            e[j] = __expf(s_G[t][j] * ATT_SCALE - m);
            sum += e[j];
        }
        const float r = 1.f / sum;
#pragma unroll
        for (int j = 0; j < 8; ++j) s_am[t][j] = e[j] * r;
    }
    __syncthreads();
    // Fold: out = w_up @ (am @ kqv) = (w_up @ am) @ kqv;  W2 = w_up @ am (48x8)
    for (int e = t; e < CIN * 8; e += 256) {
        const int c = e >> 3, o = e & 7;
        float s = 0.f;
#pragma unroll
        for (int j = 0; j < 8; ++j) s = fmaf(w_up[c * 8 + j], s_am[j][o], s);
        s_W2[c][o] = s;
    }
    __syncthreads();

    // ================= Phase 3: out = W2 @ kqv   (48 x H) =================
    for (int mt = 0; mt < 3; ++mt) {
        const int m0 = mt * 16;
        v2f a0, a1;
        a0.x = s_W2[m0 + col][krow];
        a0.y = s_W2[m0 + col][krow + 1];
        a1.x = s_W2[m0 + col][4 + krow];
        a1.y = s_W2[m0 + col][4 + krow + 1];
        for (int ht = wave; ht < HH / 16; ht += NWAVE) {
            const int h0 = ht * 16;
            v8f d = {};
            v2f b0, b1;
            b0.x = s_kqv[krow][h0 + col];
            b0.y = s_kqv[krow + 1][h0 + col];
            b1.x = s_kqv[4 + krow][h0 + col];
            b1.y = s_kqv[5 + krow][h0 + col];
            d = __builtin_amdgcn_wmma_f32_16x16x4_f32(
                false, a0, false, b0, (short)0, d, false, false);
            d = __builtin_amdgcn_wmma_f32_16x16x4_f32(
                false, a1, false, b1, (short)0, d, false, false);
            const int orow = m0 + ((lane >> 4) << 3);
            float* op = outb + (size_t)orow * HH + h0 + col;
#pragma unroll
            for (int r = 0; r < 8; ++r) op[(size_t)r * HH] = d[r];
        }
    }
}

extern "C" void kernel_launch(void* const* d_in, const int* in_sizes, int n_in,
                              void* d_out, int out_size, void* d_ws, size_t ws_size,
                              hipStream_t stream) {
    (void)in_sizes; (void)n_in; (void)out_size; (void)d_ws; (void)ws_size;
    const float* x      = (const float*)d_in[0];
    const float* w_down = (const float*)d_in[1];
    const float* w_up   = (const float*)d_in[2];
    float* out          = (float*)d_out;
    ultimus_fused<<<dim3(256), dim3(256), 0, stream>>>(x, w_down, w_up, out);
}